// AirfoilGCNN_16544214024642
// MI455X (gfx1250) — compile-verified
//
#include <hip/hip_runtime.h>
#include <math.h>

// ---------------------------------------------------------------------------
// Hierarchical GNN (3x SAGE + 3x GCN with top-k pooling) for MI455X / gfx1250.
// Dense h@W GEMMs run on the WMMA pipe (V_WMMA_F32_16X16X4_F32, f32-exact).
// Edge aggregation = atomic scatter; top-k = 4-pass radix select per graph.
// ---------------------------------------------------------------------------

typedef float v2f __attribute__((ext_vector_type(2)));
typedef float v8f __attribute__((ext_vector_type(8)));

#if defined(__AMDGCN__) && __has_builtin(__builtin_amdgcn_wmma_f32_16x16x4_f32)
#define HAVE_WMMA_F32 1
#else
#define HAVE_WMMA_F32 0
#endif

static constexpr int BG    = 4;
static constexpr int NPER0 = 65536;
static constexpr int N0    = BG * NPER0;
static constexpr int E0    = 6 * N0;

static inline int cdiv(long long a, long long b){ return (int)((a + b - 1) / b); }

// ----------------------------- utility kernels -----------------------------

__global__ void k_zero_f32(float* p, long long n){
  long long i = (long long)blockIdx.x * blockDim.x + threadIdx.x;
  if (i < n) p[i] = 0.f;
}
__global__ void k_zero_u32(unsigned* p, int n){
  int i = blockIdx.x * blockDim.x + threadIdx.x;
  if (i < n) p[i] = 0u;
}
__global__ void k_init_edges(const int* __restrict__ ei, int* __restrict__ src,
                             int* __restrict__ dst, int* __restrict__ val, int e){
  int i = blockIdx.x * blockDim.x + threadIdx.x;
  if (i < e){ src[i] = ei[i]; dst[i] = ei[e + i]; val[i] = 1; }
}
__global__ void k_extract(const float* __restrict__ x, float* __restrict__ h2, int n){
  int i = blockIdx.x * blockDim.x + threadIdx.x;
  if (i < n * 2){ int v = i >> 1, c = i & 1; h2[i] = x[v * 4 + 2 + c]; }
}

// ----------------------------- aggregation ---------------------------------

__global__ void k_scatter_deg(const int* __restrict__ dst, const int* __restrict__ val,
                              float* __restrict__ deg, int e){
  int i = blockIdx.x * blockDim.x + threadIdx.x;
  if (i < e && val[i]) atomicAdd(&deg[dst[i]], 1.0f);
}
__global__ void k_scatter_sum(const float* __restrict__ h, const int* __restrict__ src,
                              const int* __restrict__ dst, const int* __restrict__ val,
                              float* __restrict__ agg, long long total, int C){
  long long i = (long long)blockIdx.x * blockDim.x + threadIdx.x;
  if (i >= total) return;
  int e = (int)(i / C); int c = (int)(i - (long long)e * C);
  if (val[e]) atomicAdd(&agg[(long long)dst[e] * C + c], h[(long long)src[e] * C + c]);
}
__global__ void k_mean(float* __restrict__ agg, const float* __restrict__ deg,
                       long long total, int C){
  long long i = (long long)blockIdx.x * blockDim.x + threadIdx.x;
  if (i >= total) return;
  int v = (int)(i / C);
  agg[i] /= fmaxf(deg[v], 1.f);
}

// SAGE layer 1: K=2 input, scalar (too small for WMMA).
__global__ void k_sage1(const float* __restrict__ mean2, const float* __restrict__ h2,
                        const float* __restrict__ wl, const float* __restrict__ bl,
                        const float* __restrict__ wr, float* __restrict__ out, int n){
  long long i = (long long)blockIdx.x * blockDim.x + threadIdx.x;
  if (i >= (long long)n * 64) return;
  int v = (int)(i >> 6); int j = (int)(i & 63);
  float o = mean2[v * 2] * wl[j] + mean2[v * 2 + 1] * wl[64 + j] + bl[j]
          + h2[v * 2] * wr[j] + h2[v * 2 + 1] * wr[64 + j];
  out[i] = fmaxf(o, 0.f);
}

// ----------------------------- WMMA GEMM -----------------------------------
// out[n x 64] = opt_relu( A @ W1 + (B2 ? B2 @ W2 : 0) + bias ), all f32.
// One wave32 per 16-row tile; 4 col-tiles of 16; K=64 in 16 steps of 4 via
// V_WMMA_F32_16X16X4_F32. Weights staged in LDS.
// A frag (16x4 f32): lane L holds M=L&15, K={4kk+2*(L>>4), +1}.
// B frag (4x16 f32): VGPR v, lane L holds K=4kk+v+2*(L>>4), N=L&15.
// C/D (16x16 f32):  VGPR v, lane L holds M=v+8*(L>>4), N=L&15.
__launch_bounds__(256)
__global__ void k_gemm64(const float* __restrict__ A, const float* __restrict__ W1,
                         const float* __restrict__ B2, const float* __restrict__ W2,
                         const float* __restrict__ bias, float* __restrict__ out,
                         int n, int do_relu)
{
  __shared__ float sW1[64 * 64];
  __shared__ float sW2[64 * 64];
  const bool hasB = (B2 != nullptr);
  for (int i = threadIdx.x; i < 64 * 64; i += 256){
    sW1[i] = W1[i];
    sW2[i] = hasB ? W2[i] : 0.f;
  }
  __syncthreads();

  const int lane = threadIdx.x & 31;
  const int wv   = threadIdx.x >> 5;
  const int m    = lane & 15;
  const int hi   = lane >> 4;
  const int tile = blockIdx.x * 8 + wv;
  if (tile * 16 >= n) return;
  const long long rowbase = (long long)tile * 16;

  v8f acc[4];
  for (int ct = 0; ct < 4; ++ct)
    for (int j = 0; j < 8; ++j) acc[ct][j] = 0.f;

#if HAVE_WMMA_F32
  {
    const float* Ar = A + (rowbase + m) * 64;
    for (int kk = 0; kk < 16; ++kk){
      const int k0 = 4 * kk + 2 * hi;
      v2f a; a.x = Ar[k0]; a.y = Ar[k0 + 1];
      for (int ct = 0; ct < 4; ++ct){
        v2f b; b.x = sW1[k0 * 64 + ct * 16 + m];
               b.y = sW1[(k0 + 1) * 64 + ct * 16 + m];
        acc[ct] = __builtin_amdgcn_wmma_f32_16x16x4_f32(
            false, a, false, b, (short)0, acc[ct], false, false);
      }
    }
    if (hasB){
      const float* Br = B2 + (rowbase + m) * 64;
      for (int kk = 0; kk < 16; ++kk){
        const int k0 = 4 * kk + 2 * hi;
        v2f a; a.x = Br[k0]; a.y = Br[k0 + 1];
        for (int ct = 0; ct < 4; ++ct){
          v2f b; b.x = sW2[k0 * 64 + ct * 16 + m];
                 b.y = sW2[(k0 + 1) * 64 + ct * 16 + m];
          acc[ct] = __builtin_amdgcn_wmma_f32_16x16x4_f32(
              false, a, false, b, (short)0, acc[ct], false, false);
        }
      }
    }
  }
#else
  // Scalar fallback (also used by the host semantic-check pass): computes the
  // exact same C-fragment elements this lane owns.
  for (int v = 0; v < 8; ++v){
    const long long r = rowbase + v + 8 * hi;
    for (int ct = 0; ct < 4; ++ct){
      const int col = ct * 16 + m;
      float s = 0.f;
      for (int k = 0; k < 64; ++k) s += A[r * 64 + k] * sW1[k * 64 + col];
      if (hasB) for (int k = 0; k < 64; ++k) s += B2[r * 64 + k] * sW2[k * 64 + col];
      acc[ct][v] = s;
    }
  }
#endif

  for (int ct = 0; ct < 4; ++ct){
    const int col = ct * 16 + m;
    const float bv = bias ? bias[col] : 0.f;
    for (int v = 0; v < 8; ++v){
      float val = acc[ct][v] + bv;
      if (do_relu) val = fmaxf(val, 0.f);
      out[(rowbase + v + 8 * hi) * 64 + col] = val;
    }
  }
}

// ----------------------------- GCN pieces ----------------------------------

__global__ void k_gcn_scatter(const float* __restrict__ xw, const int* __restrict__ src,
                              const int* __restrict__ dst, const int* __restrict__ val,
                              const float* __restrict__ deg, float* __restrict__ acc,
                              long long total){
  long long i = (long long)blockIdx.x * blockDim.x + threadIdx.x;
  if (i >= total) return;
  int e = (int)(i >> 6); int c = (int)(i & 63);
  if (!val[e]) return;
  int s = src[e], d = dst[e];
  float norm = rsqrtf(deg[s] + 1.f) * rsqrtf(deg[d] + 1.f);
  atomicAdd(&acc[(long long)d * 64 + c], xw[(long long)s * 64 + c] * norm);
}
__global__ void k_gcn_combine(float* __restrict__ acc, const float* __restrict__ xw,
                              const float* __restrict__ deg, const float* __restrict__ b,
                              long long total){
  long long i = (long long)blockIdx.x * blockDim.x + threadIdx.x;
  if (i >= total) return;
  int v = (int)(i >> 6); int c = (int)(i & 63);
  float val = acc[i] + xw[i] / (deg[v] + 1.f) + b[c];
  acc[i] = fmaxf(val, 0.f);
}

// ----------------------------- top-k pooling -------------------------------

__global__ void k_score(const float* __restrict__ h, const float* __restrict__ w,
                        float* __restrict__ score, unsigned* __restrict__ key, int n){
  int v = blockIdx.x * blockDim.x + threadIdx.x;
  if (v >= n) return;
  float dot = 0.f, nw = 0.f;
  for (int c = 0; c < 64; ++c){ float wc = w[c]; dot += h[(long long)v * 64 + c] * wc; nw += wc * wc; }
  float s = tanhf(dot * rsqrtf(nw));
  score[v] = s;
  unsigned u = __float_as_uint(s);
  key[v] = (u & 0x80000000u) ? ~u : (u | 0x80000000u);  // order-preserving key
}
__global__ void k_radix_init(unsigned* pref, int* kneed, int knew){
  int g = threadIdx.x;
  if (g < BG){ pref[g] = 0u; kneed[g] = knew; }
}
__global__ void k_radix_hist(const unsigned* __restrict__ key, unsigned* __restrict__ hist,
                             const unsigned* __restrict__ pref, int pass, int n, int nper){
  int v = blockIdx.x * blockDim.x + threadIdx.x;
  if (v >= n) return;
  int g = v / nper;
  unsigned kv = key[v];
  bool match = (pass == 3) || ((kv >> (8 * (pass + 1))) == pref[g]);
  if (match) atomicAdd(&hist[g * 256 + ((kv >> (8 * pass)) & 255u)], 1u);
}
__global__ void k_radix_select(const unsigned* __restrict__ hist, unsigned* __restrict__ pref,
                               int* __restrict__ kneed, int pass){
  if (threadIdx.x != 0) return;
  int g = blockIdx.x;
  int need = kneed[g];
  unsigned cum = 0; int chosen = 0;
  for (int b = 255; b >= 0; --b){
    unsigned c = hist[g * 256 + b];
    if (cum + c >= (unsigned)need){ chosen = b; break; }
    cum += c;
  }
  pref[g]  = (pass == 3) ? (unsigned)chosen : ((pref[g] << 8) | (unsigned)chosen);
  kneed[g] = need - (int)cum;    // after pass 0: #ties of the threshold to keep
}
__global__ void k_compact(const float* __restrict__ h, const float* __restrict__ score,
                          const unsigned* __restrict__ key, const unsigned* __restrict__ thresh,
                          const int* __restrict__ nties, int* __restrict__ mapping,
                          float* __restrict__ hout, unsigned* __restrict__ selc,
                          unsigned* __restrict__ tiec, int n, int nper, int knew){
  int v = blockIdx.x * blockDim.x + threadIdx.x;
  if (v >= n) return;
  int g = v / nper;
  unsigned kv = key[v], T = thresh[g];
  bool take = false;
  if (kv > T) take = true;
  else if (kv == T){ unsigned t = atomicAdd(&tiec[g], 1u); take = ((int)t < nties[g]); }
  if (take){
    unsigned p = atomicAdd(&selc[g], 1u);
    long long np = (long long)g * knew + p;
    mapping[v] = (int)np;
    float sc = score[v];
    const float* hr = h + (long long)v * 64;
    float* ho = hout + np * 64;
    for (int c = 0; c < 64; ++c) ho[c] = hr[c] * sc;   // x_new = x * tanh-score
  } else mapping[v] = -1;
}
__global__ void k_remap(int* __restrict__ src, int* __restrict__ dst, int* __restrict__ val,
                        const int* __restrict__ mapping, int e){
  int i = blockIdx.x * blockDim.x + threadIdx.x;
  if (i >= e) return;
  int s = src[i], d = dst[i];
  int ns = mapping[s], nd = mapping[d];
  int nv = (val[i] && ns >= 0 && nd >= 0) ? 1 : 0;
  src[i] = ns < 0 ? 0 : ns;
  dst[i] = nd < 0 ? 0 : nd;
  val[i] = nv;
}

// ----------------------------- readout + MLP -------------------------------

__launch_bounds__(256)
__global__ void k_readout(const float* __restrict__ h, float* __restrict__ racc, int kper){
  __shared__ float smax[256], ssum[256];
  const int g = blockIdx.x >> 6;
  const int c = blockIdx.x & 63;
  const float* basep = h + (long long)g * kper * 64;
  float m = -1e30f, s = 0.f;
  for (int i = threadIdx.x; i < kper; i += 256){
    float v = basep[(long long)i * 64 + c];
    m = fmaxf(m, v); s += v;
  }
  smax[threadIdx.x] = m; ssum[threadIdx.x] = s;
  __syncthreads();
  for (int w = 128; w > 0; w >>= 1){
    if (threadIdx.x < w){
      smax[threadIdx.x] = fmaxf(smax[threadIdx.x], smax[threadIdx.x + w]);
      ssum[threadIdx.x] += ssum[threadIdx.x + w];
    }
    __syncthreads();
  }
  if (threadIdx.x == 0){
    racc[g * 128 + c]      += smax[0];
    racc[g * 128 + 64 + c] += ssum[0] / (float)kper;
  }
}

__global__ void k_mlp(const float* __restrict__ racc,
                      const float* __restrict__ w1, const float* __restrict__ b1,
                      const float* __restrict__ w2, const float* __restrict__ b2,
                      const float* __restrict__ w3, const float* __restrict__ b3,
                      float* __restrict__ out){
  __shared__ float z1[128];
  __shared__ float z2[64];
  const int t = threadIdx.x;
  for (int g = 0; g < BG; ++g){
    float s = b1[t];
    for (int i = 0; i < 128; ++i) s += racc[g * 128 + i] * w1[i * 128 + t];
    z1[t] = fmaxf(s, 0.f);
    __syncthreads();
    if (t < 64){
      float s2 = b2[t];
      for (int i = 0; i < 128; ++i) s2 += z1[i] * w2[i * 64 + t];
      z2[t] = fmaxf(s2, 0.f);
    }
    __syncthreads();
    if (t == 0){
      float s3 = b3[0];
      for (int i = 0; i < 64; ++i) s3 += z2[i] * w3[i];
      out[g] = s3;
    }
    __syncthreads();
  }
}

// ----------------------------- host orchestration --------------------------

extern "C" void kernel_launch(void* const* d_in, const int* in_sizes, int n_in,
                              void* d_out, int out_size, void* d_ws, size_t ws_size,
                              hipStream_t stream)
{
  (void)in_sizes; (void)n_in; (void)out_size; (void)ws_size;
  const float* x   = (const float*)d_in[0];
  const int*   ei  = (const int*)  d_in[1];
  const float* s_wl[3] = { (const float*)d_in[2], (const float*)d_in[5], (const float*)d_in[8] };
  const float* s_bl[3] = { (const float*)d_in[3], (const float*)d_in[6], (const float*)d_in[9] };
  const float* s_wr[3] = { (const float*)d_in[4], (const float*)d_in[7], (const float*)d_in[10] };
  const float* g_w[3]  = { (const float*)d_in[11], (const float*)d_in[13], (const float*)d_in[15] };
  const float* g_b[3]  = { (const float*)d_in[12], (const float*)d_in[14], (const float*)d_in[16] };
  const float* poolw = (const float*)d_in[17];
  const float* l1w = (const float*)d_in[18], *l1b = (const float*)d_in[19];
  const float* l2w = (const float*)d_in[20], *l2b = (const float*)d_in[21];
  const float* l3w = (const float*)d_in[22], *l3b = (const float*)d_in[23];
  float* out = (float*)d_out;

  char* base = (char*)d_ws; size_t off = 0;
  auto alloc = [&](size_t bytes) -> void* {
    off = (off + 255) & ~(size_t)255;
    void* p = base + off; off += bytes; return p;
  };
  float*    hA      = (float*)   alloc((size_t)N0 * 64 * 4);
  float*    hB      = (float*)   alloc((size_t)N0 * 64 * 4);
  float*    h2      = (float*)   alloc((size_t)N0 * 2 * 4);
  float*    m2      = (float*)   alloc((size_t)N0 * 2 * 4);
  float*    deg     = (float*)   alloc((size_t)N0 * 4);
  float*    score   = (float*)   alloc((size_t)N0 * 4);
  unsigned* key     = (unsigned*)alloc((size_t)N0 * 4);
  int*      mapping = (int*)     alloc((size_t)N0 * 4);
  int*      esrc    = (int*)     alloc((size_t)E0 * 4);
  int*      edst    = (int*)     alloc((size_t)E0 * 4);
  int*      eval    = (int*)     alloc((size_t)E0 * 4);
  unsigned* hist    = (unsigned*)alloc((size_t)BG * 256 * 4);
  unsigned* pref    = (unsigned*)alloc((size_t)BG * 4);
  int*      kneed   = (int*)     alloc((size_t)BG * 4);
  unsigned* selc    = (unsigned*)alloc((size_t)BG * 4);
  unsigned* tiec    = (unsigned*)alloc((size_t)BG * 4);
  float*    racc    = (float*)   alloc((size_t)BG * 128 * 4);

  int n = N0, nper = NPER0;

  k_zero_f32 <<<cdiv(BG * 128, 256), 256, 0, stream>>>(racc, BG * 128);
  k_init_edges<<<cdiv(E0, 256), 256, 0, stream>>>(ei, esrc, edst, eval, E0);
  k_extract  <<<cdiv((long long)N0 * 2, 256), 256, 0, stream>>>(x, h2, N0);

  auto pool = [&](const float* hin, float* hout, int layer){
    int knew = nper / 2;
    k_score<<<cdiv(n, 256), 256, 0, stream>>>(hin, poolw + layer * 64, score, key, n);
    k_radix_init<<<1, 32, 0, stream>>>(pref, kneed, knew);
    for (int pass = 3; pass >= 0; --pass){
      k_zero_u32<<<cdiv(BG * 256, 256), 256, 0, stream>>>(hist, BG * 256);
      k_radix_hist<<<cdiv(n, 256), 256, 0, stream>>>(key, hist, pref, pass, n, nper);
      k_radix_select<<<BG, 32, 0, stream>>>(hist, pref, kneed, pass);
    }
    k_zero_u32<<<1, 32, 0, stream>>>(selc, BG);
    k_zero_u32<<<1, 32, 0, stream>>>(tiec, BG);
    k_compact<<<cdiv(n, 256), 256, 0, stream>>>(hin, score, key, pref, kneed,
                                                mapping, hout, selc, tiec, n, nper, knew);
    k_remap<<<cdiv(E0, 256), 256, 0, stream>>>(esrc, edst, eval, mapping, E0);
    k_readout<<<BG * 64, 256, 0, stream>>>(hout, racc, knew);
    nper = knew; n = BG * knew;
  };

  // -------- SAGE layer 1 (input width 2) --------
  k_zero_f32<<<cdiv((long long)n * 2, 256), 256, 0, stream>>>(m2, (long long)n * 2);
  k_zero_f32<<<cdiv(n, 256), 256, 0, stream>>>(deg, n);
  k_scatter_deg<<<cdiv(E0, 256), 256, 0, stream>>>(edst, eval, deg, E0);
  k_scatter_sum<<<cdiv((long long)E0 * 2, 256), 256, 0, stream>>>(h2, esrc, edst, eval, m2,
                                                                  (long long)E0 * 2, 2);
  k_mean<<<cdiv((long long)n * 2, 256), 256, 0, stream>>>(m2, deg, (long long)n * 2, 2);
  k_sage1<<<cdiv((long long)n * 64, 256), 256, 0, stream>>>(m2, h2, s_wl[0], s_bl[0], s_wr[0], hA, n);
  pool(hA, hB, 0);
  float* cur = hB; float* oth = hA;

  // -------- SAGE layers 2,3 (width 64, WMMA) --------
  for (int L = 1; L < 3; ++L){
    k_zero_f32<<<cdiv((long long)n * 64, 256), 256, 0, stream>>>(oth, (long long)n * 64);
    k_zero_f32<<<cdiv(n, 256), 256, 0, stream>>>(deg, n);
    k_scatter_deg<<<cdiv(E0, 256), 256, 0, stream>>>(edst, eval, deg, E0);
    k_scatter_sum<<<cdiv((long long)E0 * 64, 256), 256, 0, stream>>>(cur, esrc, edst, eval, oth,
                                                                     (long long)E0 * 64, 64);
    k_mean<<<cdiv((long long)n * 64, 256), 256, 0, stream>>>(oth, deg, (long long)n * 64, 64);
    // out = relu(mean@wl + h@wr + bl); out aliases A-input (safe: per-wave tile)
    k_gemm64<<<n / 128, 256, 0, stream>>>(oth, s_wl[L], cur, s_wr[L], s_bl[L], oth, n, 1);
    pool(oth, cur, L);
  }

  // -------- GCN layers 4..6 (width 64, WMMA) --------
  for (int L = 0; L < 3; ++L){
    k_zero_f32<<<cdiv(n, 256), 256, 0, stream>>>(deg, n);
    k_scatter_deg<<<cdiv(E0, 256), 256, 0, stream>>>(edst, eval, deg, E0);
    k_gemm64<<<n / 128, 256, 0, stream>>>(cur, g_w[L], nullptr, nullptr, nullptr, oth, n, 0); // xw
    k_zero_f32<<<cdiv((long long)n * 64, 256), 256, 0, stream>>>(cur, (long long)n * 64);
    k_gcn_scatter<<<cdiv((long long)E0 * 64, 256), 256, 0, stream>>>(oth, esrc, edst, eval, deg, cur,
                                                                     (long long)E0 * 64);
    k_gcn_combine<<<cdiv((long long)n * 64, 256), 256, 0, stream>>>(cur, oth, deg, g_b[L],
                                                                    (long long)n * 64);
    pool(cur, oth, 3 + L);
    { float* t = cur; cur = oth; oth = t; }
  }

  // -------- MLP head --------
  k_mlp<<<1, 128, 0, stream>>>(racc, l1w, l1b, l2w, l2b, l3w, l3b, out);
}